// local_attention_extration_50044958933190
// MI455X (gfx1250) — compile-verified
//
#include <hip/hip_runtime.h>
#include <math.h>

typedef __attribute__((ext_vector_type(2))) float v2f;
typedef __attribute__((ext_vector_type(8))) float v8f;

#define NPTS 4096
#define KNN  20
#define ROWS 64          // rows (points) per block
#define BLK  64          // 2 wave32

// D = A(16x4) * B(4x16) + C, fp32. A slot K = (x, y, z, 1) for row n,
// B slot K = (2x, 2y, 2z, -|x|^2) for col m  ->  D = 2<xn,xm> - |xm|^2.
// Per-row top-k ranking of this equals ranking of -dist^2 (row term constant).
__device__ __forceinline__ v8f wmma_f32_16x16x4(v2f a, v2f b, v8f c) {
    return __builtin_amdgcn_wmma_f32_16x16x4_f32(
        /*neg_a=*/false, a, /*neg_b=*/false, b,
        /*c_mod=*/(short)0, c, /*reuse_a=*/false, /*reuse_b=*/false);
}

__global__ __launch_bounds__(BLK) void knn_attn_kernel(
    const float* __restrict__ g_x,                       // [B,3,N]
    const float* __restrict__ g_w1, const float* __restrict__ g_b1,
    const float* __restrict__ g_w2, const float* __restrict__ g_b2,
    const float* __restrict__ g_w3, const float* __restrict__ g_b3,
    float* __restrict__ g_out)                           // [B,16,N]
{
    __shared__ float s_x[NPTS], s_y[NPTS], s_z[NPTS];    // 48 KB coords of batch
    __shared__ float s_tb[ROWS * 17];                    // 16x16 negdist tiles (padded)
    __shared__ float s_kd[KNN * ROWS];                   // per-row top-k scores (desc)
    __shared__ unsigned short s_ki[KNN * ROWS];          // per-row top-k indices
    __shared__ float s_w1[48], s_b1[16], s_w2[48], s_b2[16], s_w3[16];
    __shared__ float s_b3;

    const int tid      = threadIdx.x;
    const int b        = blockIdx.x / (NPTS / ROWS);
    const int rowBase  = (blockIdx.x % (NPTS / ROWS)) * ROWS;
    const float* xb    = g_x + (size_t)b * 3 * NPTS;

    // ---- stage batch coords + weights into LDS ----
    for (int m = tid; m < NPTS; m += BLK) {
        s_x[m] = xb[m];
        s_y[m] = xb[NPTS + m];
        s_z[m] = xb[2 * NPTS + m];
    }
    if (tid < 48) { s_w1[tid] = g_w1[tid]; s_w2[tid] = g_w2[tid]; }
    if (tid < 16) { s_b1[tid] = g_b1[tid]; s_b2[tid] = g_b2[tid]; s_w3[tid] = g_w3[tid]; }
    if (tid == 0) { s_b3 = g_b3[0]; }
#pragma unroll
    for (int k = 0; k < KNN; ++k) {
        s_kd[k * ROWS + tid] = -3.0e38f;
        s_ki[k * ROWS + tid] = 0;
    }
    __syncthreads();

    const int  lid    = tid & 31;
    const int  wv     = tid >> 5;         // wave 0/1, each owns 32 rows (2 tiles)
    const int  lane16 = lid & 15;
    const bool hi     = (lid >= 16);

    // ---- pack A (16x4 f32): vgpr0 = K0(lanes0-15)|K2(lanes16-31), vgpr1 = K1|K3 ----
    v2f at0, at1;
    {
        const int r0 = rowBase + wv * 32 + lane16;        // tile 0 rows
        const int r1 = r0 + 16;                           // tile 1 rows
        at0.x = hi ? s_z[r0] : s_x[r0];
        at0.y = hi ? 1.0f    : s_y[r0];
        at1.x = hi ? s_z[r1] : s_x[r1];
        at1.y = hi ? 1.0f    : s_y[r1];
    }
    const int rowOff = wv * 32 + (hi ? 8 : 0);

    // ---- stream 16-column chunks: WMMA score tiles -> LDS -> per-row top-k ----
    for (int cc = 0; cc < NPTS; cc += 16) {
        // pack B (4x16 f32): K = (2x, 2y, 2z, -|x|^2) of column point
        const int   col = cc + lane16;
        const float cx = s_x[col], cy = s_y[col], cz = s_z[col];
        const float nxx = -(cx * cx + cy * cy + cz * cz);
        v2f bv;
        bv.x = hi ? (2.0f * cz) : (2.0f * cx);
        bv.y = hi ? nxx         : (2.0f * cy);

        v8f c0 = {};
        v8f c1 = {};
        c0 = wmma_f32_16x16x4(at0, bv, c0);
        c1 = wmma_f32_16x16x4(at1, bv, c1);

        // C layout: vgpr r, lanes0-15 -> M=r, N=lane; lanes16-31 -> M=r+8, N=lane-16
#pragma unroll
        for (int r = 0; r < 8; ++r) {
            s_tb[(rowOff + r)      * 17 + lane16] = c0[r];
            s_tb[(rowOff + 16 + r) * 17 + lane16] = c1[r];
        }
        __syncthreads();

        // thread tid owns block-row tid: insert 16 candidates into sorted top-20
        float worst = s_kd[(KNN - 1) * ROWS + tid];
#pragma unroll 1
        for (int j = 0; j < 16; ++j) {
            const float v = s_tb[tid * 17 + j];
            if (v > worst) {
                int p = KNN - 1;
                while (p > 0 && s_kd[(p - 1) * ROWS + tid] < v) {
                    s_kd[p * ROWS + tid] = s_kd[(p - 1) * ROWS + tid];
                    s_ki[p * ROWS + tid] = s_ki[(p - 1) * ROWS + tid];
                    --p;
                }
                s_kd[p * ROWS + tid] = v;
                s_ki[p * ROWS + tid] = (unsigned short)(cc + j);
                worst = s_kd[(KNN - 1) * ROWS + tid];
            }
        }
        __syncthreads();
    }

    // ---- phase 2: fused edge-conv + attention + ELU for point n = rowBase+tid ----
    const int   n  = rowBase + tid;
    const float px = s_x[n], py = s_y[n], pz = s_z[n];

    // sa = w3 . (w1 @ p + b1) + b3
    float sa = s_b3;
#pragma unroll
    for (int o = 0; o < 16; ++o) {
        const float f = s_b1[o] + s_w1[3 * o] * px + s_w1[3 * o + 1] * py + s_w1[3 * o + 2] * pz;
        sa += s_w3[o] * f;
    }

    float lg[KNN];
    float mx = -3.0e38f;
#pragma unroll
    for (int k = 0; k < KNN; ++k) {
        const int m = (int)s_ki[k * ROWS + tid];
        const float dx = px - s_x[m], dy = py - s_y[m], dz = pz - s_z[m];
        float na = s_b3;
#pragma unroll
        for (int o = 0; o < 16; ++o) {
            const float v = s_b2[o] + s_w2[3 * o] * dx + s_w2[3 * o + 1] * dy + s_w2[3 * o + 2] * dz;
            na += s_w3[o] * v;
        }
        float l = sa + na;
        l = (l > 0.0f) ? l : 0.01f * l;   // leaky_relu(0.01)
        lg[k] = l;
        mx = fmaxf(mx, l);
    }
    float ssum = 0.0f;
#pragma unroll
    for (int k = 0; k < KNN; ++k) { lg[k] = __expf(lg[k] - mx); ssum += lg[k]; }
    const float inv = 1.0f / ssum;

    float vals[16];
#pragma unroll
    for (int o = 0; o < 16; ++o) vals[o] = 0.0f;
#pragma unroll
    for (int k = 0; k < KNN; ++k) {
        const float coef = lg[k] * inv;
        const int m = (int)s_ki[k * ROWS + tid];
        const float dx = px - s_x[m], dy = py - s_y[m], dz = pz - s_z[m];
#pragma unroll
        for (int o = 0; o < 16; ++o) {
            const float v = s_b2[o] + s_w2[3 * o] * dx + s_w2[3 * o + 1] * dy + s_w2[3 * o + 2] * dz;
            vals[o] += coef * v;
        }
    }

    float* ob = g_out + (size_t)b * 16 * NPTS;
#pragma unroll
    for (int o = 0; o < 16; ++o) {
        float r = vals[o];
        r = (r > 0.0f) ? r : (__expf(r) - 1.0f);       // elu
        ob[o * NPTS + n] = r;
    }
}

extern "C" void kernel_launch(void* const* d_in, const int* in_sizes, int n_in,
                              void* d_out, int out_size, void* d_ws, size_t ws_size,
                              hipStream_t stream) {
    const float* x  = (const float*)d_in[0];
    const float* w1 = (const float*)d_in[1];
    const float* b1 = (const float*)d_in[2];
    const float* w2 = (const float*)d_in[3];
    const float* b2 = (const float*)d_in[4];
    const float* w3 = (const float*)d_in[5];
    const float* b3 = (const float*)d_in[6];
    float* out = (float*)d_out;

    const int B = in_sizes[0] / (3 * NPTS);
    dim3 grid(B * (NPTS / ROWS));
    knn_attn_kernel<<<grid, BLK, 0, stream>>>(x, w1, b1, w2, b2, w3, b3, out);
}